// EncoderGNN_47665547051053
// MI455X (gfx1250) — compile-verified
//
#include <hip/hip_runtime.h>
#include <string.h>

typedef float v2f __attribute__((ext_vector_type(2)));
typedef float v8f __attribute__((ext_vector_type(8)));

#define N_NODES 10000
#define N_EDGES 160000
#define SDIM 128
#define VDIM 32
#define RBF_DIM 32
#define KIN 288          // 2*SDIM + RBF_DIM
#define PI_F 3.14159265358979323846f

// ---------------------------------------------------------------------------
// fp32 hardware atomic add
// ---------------------------------------------------------------------------
__device__ __forceinline__ void atomAddF(float* p, float v) {
    unsafeAtomicAdd(p, v);   // global_atomic_add_f32
}

// ---------------------------------------------------------------------------
// silu
// ---------------------------------------------------------------------------
__device__ __forceinline__ float siluf(float x) {
    return x / (1.0f + __expf(-x));
}

// ---------------------------------------------------------------------------
// async L2 -> LDS copies (CDNA5 GLOBAL_LOAD_ASYNC_TO_LDS, ASYNCcnt-tracked)
// ---------------------------------------------------------------------------
__device__ __forceinline__ void asyncLdsB128(unsigned lds_off, const void* gaddr) {
    asm volatile("global_load_async_to_lds_b128 %0, %1, off"
                 :: "v"(lds_off), "v"(gaddr) : "memory");
}
__device__ __forceinline__ void asyncLdsB32(unsigned lds_off, const void* gaddr) {
    asm volatile("global_load_async_to_lds_b32 %0, %1, off"
                 :: "v"(lds_off), "v"(gaddr) : "memory");
}
__device__ __forceinline__ void waitAsync0() {
    asm volatile("s_wait_asynccnt 0x0" ::: "memory");
}
__device__ __forceinline__ unsigned ldsOffset(const void* p) {
    // generic pointer to __shared__: low 32 bits are the wave-relative LDS offset
    return (unsigned)(uintptr_t)p;
}

// ---------------------------------------------------------------------------
// RBF + cosine-cutoff envelope (computed once; d is static across layers)
// ---------------------------------------------------------------------------
__global__ __launch_bounds__(256) void k_rbf(const float* __restrict__ d,
                                             float* __restrict__ rbf,
                                             float* __restrict__ C, int E) {
    int e = blockIdx.x * 256 + threadIdx.x;
    if (e >= E) return;
    float dd = d[e];
    float c = 0.5f * (cosf(PI_F * dd * 0.1f) + 1.0f);
    C[e] = (dd < 10.0f) ? c : 0.0f;
    const float inv_sigma = (float)RBF_DIM / 10.0f;   // 1/sigma = 3.2
    #pragma unroll
    for (int j = 0; j < RBF_DIM; ++j) {
        float mu = (10.0f / 31.0f) * (float)j;
        float t = (dd - mu) * inv_sigma;
        rbf[(size_t)e * RBF_DIM + j] = expf(-0.5f * t * t);
    }
}

__global__ __launch_bounds__(256) void k_cnt(const long long* __restrict__ dst,
                                             float* __restrict__ cnt, int E) {
    int e = blockIdx.x * 256 + threadIdx.x;
    if (e < E) atomAddF(&cnt[(int)dst[e]], 1.0f);
}

__global__ __launch_bounds__(256) void k_invcnt(float* __restrict__ cnt, int n) {
    int i = blockIdx.x * 256 + threadIdx.x;
    if (i < n) cnt[i] = 1.0f / fmaxf(cnt[i], 1.0f);
}

// ---------------------------------------------------------------------------
// Equivariant layernorm: one wave32 per node (8 nodes / 256-thread block)
// ---------------------------------------------------------------------------
__global__ __launch_bounds__(256) void k_layernorm(const float* __restrict__ s,
                                                   const float* __restrict__ v,
                                                   const float* __restrict__ lw,
                                                   const float* __restrict__ lb,
                                                   float* __restrict__ sn,
                                                   float* __restrict__ vn) {
    const int wid  = threadIdx.x >> 5;
    const int lane = threadIdx.x & 31;
    const int nd   = blockIdx.x * 8 + wid;
    const float* sp = s + (size_t)nd * SDIM;

    float x0 = sp[lane], x1 = sp[lane + 32], x2 = sp[lane + 64], x3 = sp[lane + 96];
    float sum = x0 + x1 + x2 + x3;
    #pragma unroll
    for (int m = 16; m >= 1; m >>= 1) sum += __shfl_xor(sum, m, 32);
    float mu = sum * (1.0f / 128.0f);
    float d0 = x0 - mu, d1 = x1 - mu, d2 = x2 - mu, d3 = x3 - mu;
    float vs = d0 * d0 + d1 * d1 + d2 * d2 + d3 * d3;
    #pragma unroll
    for (int m = 16; m >= 1; m >>= 1) vs += __shfl_xor(vs, m, 32);
    float rinv = rsqrtf(vs * (1.0f / 128.0f) + 1e-6f);
    float* so = sn + (size_t)nd * SDIM;
    so[lane]      = d0 * rinv * lw[lane]      + lb[lane];
    so[lane + 32] = d1 * rinv * lw[lane + 32] + lb[lane + 32];
    so[lane + 64] = d2 * rinv * lw[lane + 64] + lb[lane + 64];
    so[lane + 96] = d3 * rinv * lw[lane + 96] + lb[lane + 96];

    const float* vp = v + (size_t)nd * 3 * VDIM;
    float a0 = vp[lane], a1 = vp[lane + 32], a2 = vp[lane + 64];
    float ss = a0 * a0 + a1 * a1 + a2 * a2;
    #pragma unroll
    for (int m = 16; m >= 1; m >>= 1) ss += __shfl_xor(ss, m, 32);
    float vrms = sqrtf(ss * (1.0f / 32.0f));
    float inv = 1.0f / (vrms + 1e-6f);
    float* vo = vn + (size_t)nd * 3 * VDIM;
    vo[lane] = a0 * inv; vo[lane + 32] = a1 * inv; vo[lane + 64] = a2 * inv;
}

// ---------------------------------------------------------------------------
// Generic f32 WMMA GEMM: C[M x N] = act(A[M x K] @ B[K x N] + bias)
// one wave per 16x16 tile; grid = (M/16, N/16); K % 4 == 0
// ---------------------------------------------------------------------------
__global__ __launch_bounds__(32)
void k_gemm(const float* __restrict__ A, int lda,
            const float* __restrict__ B, int ldb,
            const float* __restrict__ bias,
            float* __restrict__ C, int ldc,
            int M, int K, int act) {
    const int lane = threadIdx.x;
    const int half = lane >> 4, l15 = lane & 15;
    const int m0 = blockIdx.x * 16;
    const int n  = blockIdx.y * 16 + l15;
    int mr = m0 + l15; if (mr >= M) mr = M - 1;
    const float* Arow = A + (size_t)mr * lda;

    v8f acc;
    float bv = bias ? bias[n] : 0.0f;
    #pragma unroll
    for (int j = 0; j < 8; ++j) acc[j] = bv;

    for (int k = 0; k < K; k += 4) {
        int ka = k + 2 * half;
        v2f a, b;
        a.x = Arow[ka];
        a.y = Arow[ka + 1];
        b.x = B[(size_t)ka * ldb + n];
        b.y = B[(size_t)(ka + 1) * ldb + n];
        acc = __builtin_amdgcn_wmma_f32_16x16x4_f32(false, a, false, b,
                                                    (short)0, acc, false, false);
    }
    if (act) {
        #pragma unroll
        for (int j = 0; j < 8; ++j) acc[j] = siluf(acc[j]);
    }
    #pragma unroll
    for (int j = 0; j < 8; ++j) {
        int m = m0 + j + 8 * half;
        if (m < M) C[(size_t)m * ldc + n] = acc[j];
    }
}

// ---------------------------------------------------------------------------
// Fused edge kernel: async-gather X=[s_dst|s_src|rbf] into LDS,
// H=silu(X@We1+be1), OUT=H@We2+be2, scatter sm/vm via f32 atomics.
// 32 edges per 256-thread (8-wave) block; both GEMMs use f32 WMMA.
// ---------------------------------------------------------------------------
template <int GV>
__global__ __launch_bounds__(256)
void k_edge(const float* __restrict__ sn,  const float* __restrict__ sWs,
            const float* __restrict__ vWv,
            const float* __restrict__ rbf, const float* __restrict__ Cb,
            const float* __restrict__ rvec,
            const long long* __restrict__ src, const long long* __restrict__ dst,
            const float* __restrict__ We1, const float* __restrict__ be1,
            const float* __restrict__ We2, const float* __restrict__ be2,
            float* __restrict__ s_agg, float* __restrict__ v_agg, int E) {
    constexpr int TM  = 32;
    constexpr int HID = 128;
    constexpr int N2  = 128 + GV;
    constexpr int NT2 = N2 / 16;

    __shared__ float Xs[TM][KIN + 4];
    __shared__ float Hs[TM][HID + 4];
    __shared__ float Gv[TM][GV + 4];
    __shared__ int   eS[TM];
    __shared__ int   eD[TM];
    __shared__ float eC[TM];

    const int tid  = threadIdx.x;
    const int lane = tid & 31;
    const int w    = tid >> 5;
    const int half = lane >> 4;
    const int l15  = lane & 15;
    const int e0   = blockIdx.x * TM;

    if (tid < TM) {
        int e = e0 + tid; int ec = (e < E) ? e : (E - 1);
        eS[tid] = (int)src[ec];
        eD[tid] = (int)dst[ec];
        eC[tid] = (e < E) ? Cb[ec] : 0.0f;
    }
    __syncthreads();

    // --- async gather of the X tile: 4 rows per wave, direct L2 -> LDS ---
    {
        #pragma unroll
        for (int r = 0; r < 4; ++r) {
            const int m  = w * 4 + r;
            const int e  = e0 + m;
            const int ec = (e < E) ? e : (E - 1);
            // s[dst] segment: 512B = 32 lanes x 16B
            asyncLdsB128(ldsOffset(&Xs[m][lane * 4]),
                         sn + (size_t)eD[m] * SDIM + lane * 4);
            // s[src] segment
            asyncLdsB128(ldsOffset(&Xs[m][128 + lane * 4]),
                         sn + (size_t)eS[m] * SDIM + lane * 4);
            // rbf segment: 128B = 32 lanes x 4B
            asyncLdsB32(ldsOffset(&Xs[m][256 + lane]),
                        rbf + (size_t)ec * RBF_DIM + lane);
        }
        waitAsync0();
    }
    __syncthreads();

    // stage A: H = silu(X @ We1 + be1)   (2 row-tiles x 8 col-tiles)
    for (int t = w; t < 16; t += 8) {
        int rt = t >> 3, ct = t & 7;
        int n  = ct * 16 + l15;
        int mr = rt * 16 + l15;
        v8f acc; float bv = be1[n];
        #pragma unroll
        for (int j = 0; j < 8; ++j) acc[j] = bv;
        for (int k = 0; k < KIN; k += 4) {
            int ka = k + 2 * half;
            v2f a, b;
            a.x = Xs[mr][ka];
            a.y = Xs[mr][ka + 1];
            b.x = We1[(size_t)ka * HID + n];
            b.y = We1[(size_t)(ka + 1) * HID + n];
            acc = __builtin_amdgcn_wmma_f32_16x16x4_f32(false, a, false, b,
                                                        (short)0, acc, false, false);
        }
        #pragma unroll
        for (int j = 0; j < 8; ++j)
            Hs[rt * 16 + j + 8 * half][n] = siluf(acc[j]);
    }
    __syncthreads();

    // stage B: OUT = H @ We2 + be2; gs -> sm atomic scatter, gv -> LDS
    for (int t = w; t < 2 * NT2; t += 8) {
        int rt = t & 1, ct = t >> 1;
        int n  = ct * 16 + l15;
        int mr = rt * 16 + l15;
        v8f acc; float bv = be2[n];
        #pragma unroll
        for (int j = 0; j < 8; ++j) acc[j] = bv;
        for (int k = 0; k < HID; k += 4) {
            int ka = k + 2 * half;
            v2f a, b;
            a.x = Hs[mr][ka];
            a.y = Hs[mr][ka + 1];
            b.x = We2[(size_t)ka * N2 + n];
            b.y = We2[(size_t)(ka + 1) * N2 + n];
            acc = __builtin_amdgcn_wmma_f32_16x16x4_f32(false, a, false, b,
                                                        (short)0, acc, false, false);
        }
        if (ct < 8) {   // gs columns: sm = C * gs * sWs[src], scatter to dst
            #pragma unroll
            for (int j = 0; j < 8; ++j) {
                int m = rt * 16 + j + 8 * half;
                float val = eC[m] * acc[j] * sWs[(size_t)eS[m] * SDIM + n];
                if (e0 + m < E)
                    atomAddF(&s_agg[(size_t)eD[m] * SDIM + n], val);
            }
        } else {        // gv columns -> LDS for vector-message epilogue
            #pragma unroll
            for (int j = 0; j < 8; ++j)
                Gv[rt * 16 + j + 8 * half][n - 128] = acc[j];
        }
    }
    __syncthreads();

    // vm epilogue: vm = C * (gv1 * r  [+ gv2 * (v@Wv_in)[src]])
    for (int idx = tid; idx < TM * 3 * VDIM; idx += 256) {
        int m = idx / 96, rem = idx % 96;
        int kk = rem >> 5, c = rem & 31;
        int e = e0 + m; int ec = (e < E) ? e : (E - 1);
        float val = Gv[m][c] * rvec[(size_t)ec * 3 + kk];
        if (GV == 64)
            val += Gv[m][32 + c] * vWv[((size_t)eS[m] * 3 + kk) * VDIM + c];
        val *= eC[m];
        if (e < E)
            atomAddF(&v_agg[((size_t)eD[m] * 3 + kk) * VDIM + c], val);
    }
}

// ---------------------------------------------------------------------------
// s_cur = sn + s_agg ; v_cur = vn + v_agg / max(cnt,1)
// ---------------------------------------------------------------------------
__global__ __launch_bounds__(256)
void k_combine(const float* __restrict__ sn, const float* __restrict__ s_agg,
               const float* __restrict__ vn, const float* __restrict__ v_agg,
               const float* __restrict__ invc,
               float* __restrict__ s_cur, float* __restrict__ v_cur) {
    int i = blockIdx.x * 256 + threadIdx.x;
    if (i < N_NODES * SDIM) {
        s_cur[i] = sn[i] + s_agg[i];
    } else {
        int j = i - N_NODES * SDIM;
        if (j < N_NODES * 3 * VDIM)
            v_cur[j] = vn[j] + v_agg[j] * invc[j / 96];
    }
}

// ---------------------------------------------------------------------------
// U_in = [s | sqrt(sum_k vmix[:,k,32:64]^2 + eps)]
// ---------------------------------------------------------------------------
__global__ __launch_bounds__(256)
void k_uin(const float* __restrict__ s_cur, const float* __restrict__ vmix,
           float* __restrict__ U) {
    int i = blockIdx.x * 256 + threadIdx.x;
    if (i >= N_NODES * (SDIM + VDIM)) return;
    int nd = i / 160, c = i % 160;
    if (c < 128) {
        U[i] = s_cur[(size_t)nd * SDIM + c];
    } else {
        int cc = c - 128;
        float acc = 1e-6f;
        #pragma unroll
        for (int k = 0; k < 3; ++k) {
            float x = vmix[((size_t)nd * 3 + k) * 64 + 32 + cc];
            acc += x * x;
        }
        U[i] = sqrtf(acc);
    }
}

// ---------------------------------------------------------------------------
// s += u[:, :128] ; v += u[:, 128:][None] * v_keep (vmix[:, :, :32])
// ---------------------------------------------------------------------------
__global__ __launch_bounds__(256)
void k_final(const float* __restrict__ Uout, const float* __restrict__ vmix,
             float* __restrict__ s_cur, float* __restrict__ v_cur) {
    int i = blockIdx.x * 256 + threadIdx.x;
    if (i >= N_NODES * 224) return;
    int nd = i / 224, c = i % 224;
    if (c < 128) {
        s_cur[(size_t)nd * SDIM + c] += Uout[(size_t)nd * 160 + c];
    } else {
        int t = c - 128;
        int k = t >> 5, cc = t & 31;
        v_cur[((size_t)nd * 3 + k) * VDIM + cc] +=
            Uout[(size_t)nd * 160 + 128 + cc] * vmix[((size_t)nd * 3 + k) * 64 + cc];
    }
}

// ---------------------------------------------------------------------------
// Host side
// ---------------------------------------------------------------------------
struct LayerP {
    const float *ln_w, *ln_b, *We1, *be1, *We2, *be2, *Ws, *bs, *Wv_mix;
    const float *Wv_in, *Wu1, *bu1, *Wu2, *bu2, *Wu, *bu;
    int gv;
};

extern "C" void kernel_launch(void* const* d_in, const int* in_sizes, int n_in,
                              void* d_out, int out_size, void* d_ws, size_t ws_size,
                              hipStream_t stream) {
    const int E = N_EDGES;
    const float* s_in = (const float*)d_in[0];
    const float* v_in = (const float*)d_in[1];
    const long long* ei = (const long long*)d_in[2];
    const float* dvec = (const float*)d_in[3];
    const float* rvec = (const float*)d_in[4];
    const long long* srcp = ei;
    const long long* dstp = ei + E;

    // ---- map param leaves (handles dict-insertion or sorted-key flattening)
    LayerP L[5]; memset(L, 0, sizeof(L));
    {
        int idx = 5;
        bool sorted = (n_in > 5 && in_sizes[5] != 128);  // sorted order starts with We1
        for (int i = 0; i < 5; ++i) {
            LayerP& p = L[i];
            bool has_vin = (i > 0);
            bool use_mlp = (i < 4);
            p.gv = has_vin ? 64 : 32;
            const float* q[16];
            int nl = 9 + (has_vin ? 1 : 0) + (use_mlp ? 4 : 2);
            for (int j = 0; j < nl; ++j) q[j] = (const float*)d_in[idx + j];
            idx += nl;
            if (!sorted) {
                int k = 0;
                p.ln_w = q[k++]; p.ln_b = q[k++]; p.We1 = q[k++]; p.be1 = q[k++];
                p.We2 = q[k++]; p.be2 = q[k++]; p.Ws = q[k++]; p.bs = q[k++];
                p.Wv_mix = q[k++];
                if (has_vin) p.Wv_in = q[k++];
                if (use_mlp) { p.Wu1 = q[k++]; p.bu1 = q[k++]; p.Wu2 = q[k++]; p.bu2 = q[k++]; }
                else         { p.Wu  = q[k++]; p.bu  = q[k++]; }
            } else {
                int k = 0;
                p.We1 = q[k++]; p.We2 = q[k++]; p.Ws = q[k++];
                if (use_mlp) { p.Wu1 = q[k++]; p.Wu2 = q[k++]; }
                else         { p.Wu  = q[k++]; }
                if (has_vin) p.Wv_in = q[k++];
                p.Wv_mix = q[k++];
                p.be1 = q[k++]; p.be2 = q[k++]; p.bs = q[k++];
                if (use_mlp) { p.bu1 = q[k++]; p.bu2 = q[k++]; }
                else         { p.bu  = q[k++]; }
                p.ln_b = q[k++]; p.ln_w = q[k++];
            }
        }
    }

    // ---- workspace layout (floats)
    float* W = (float*)d_ws;
    size_t off = 0;
    float* sn    = W + off; off += (size_t)N_NODES * SDIM;       // 1.28M
    float* vn    = W + off; off += (size_t)N_NODES * 3 * VDIM;   // 0.96M
    float* sWs   = W + off; off += (size_t)N_NODES * SDIM;
    float* vWv   = W + off; off += (size_t)N_NODES * 3 * VDIM;
    float* s_agg = W + off; off += (size_t)N_NODES * SDIM;
    float* v_agg = W + off; off += (size_t)N_NODES * 3 * VDIM;
    float* rbfb  = W + off; off += (size_t)E * RBF_DIM;          // 5.12M
    float* Cb    = W + off; off += (size_t)E;
    float* invc  = W + off; off += (size_t)N_NODES + 16;
    float* vmix  = W + off; off += (size_t)N_NODES * 3 * 64;     // 1.92M
    float* U_in  = W + off; off += (size_t)N_NODES * 160;
    float* Hu    = W + off; off += (size_t)N_NODES * 128;
    float* Uout  = W + off; off += (size_t)N_NODES * 160;

    float* s_cur = (float*)d_out;                     // [10000 x 128]
    float* v_cur = s_cur + (size_t)N_NODES * SDIM;    // [10000 x 3 x 32]

    // ---- one-time (per call) setup
    hipMemcpyAsync(s_cur, s_in, (size_t)N_NODES * SDIM * 4, hipMemcpyDeviceToDevice, stream);
    hipMemcpyAsync(v_cur, v_in, (size_t)N_NODES * 3 * VDIM * 4, hipMemcpyDeviceToDevice, stream);
    k_rbf<<<E / 256, 256, 0, stream>>>(dvec, rbfb, Cb, E);
    hipMemsetAsync(invc, 0, (size_t)N_NODES * 4, stream);
    k_cnt<<<E / 256, 256, 0, stream>>>(dstp, invc, E);
    k_invcnt<<<(N_NODES + 255) / 256, 256, 0, stream>>>(invc, N_NODES);

    for (int i = 0; i < 5; ++i) {
        const LayerP& p = L[i];
        k_layernorm<<<N_NODES / 8, 256, 0, stream>>>(s_cur, v_cur, p.ln_w, p.ln_b, sn, vn);
        // per-node precompute: sWs = sn @ Ws + bs ; vWv = vn @ Wv_in
        k_gemm<<<dim3(N_NODES / 16, 8), 32, 0, stream>>>(sn, 128, p.Ws, 128, p.bs,
                                                         sWs, 128, N_NODES, 128, 0);
        if (p.gv == 64)
            k_gemm<<<dim3(3 * N_NODES / 16, 2), 32, 0, stream>>>(vn, 32, p.Wv_in, 32, nullptr,
                                                                 vWv, 32, 3 * N_NODES, 32, 0);
        hipMemsetAsync(s_agg, 0, (size_t)N_NODES * SDIM * 4, stream);
        hipMemsetAsync(v_agg, 0, (size_t)N_NODES * 3 * VDIM * 4, stream);
        if (p.gv == 64)
            k_edge<64><<<E / 32, 256, 0, stream>>>(sn, sWs, vWv, rbfb, Cb, rvec, srcp, dstp,
                                                   p.We1, p.be1, p.We2, p.be2, s_agg, v_agg, E);
        else
            k_edge<32><<<E / 32, 256, 0, stream>>>(sn, sWs, nullptr, rbfb, Cb, rvec, srcp, dstp,
                                                   p.We1, p.be1, p.We2, p.be2, s_agg, v_agg, E);
        k_combine<<<(N_NODES * 224 + 255) / 256, 256, 0, stream>>>(sn, s_agg, vn, v_agg, invc,
                                                                   s_cur, v_cur);
        // vmix = v @ Wv_mix
        k_gemm<<<dim3(3 * N_NODES / 16, 4), 32, 0, stream>>>(v_cur, 32, p.Wv_mix, 64, nullptr,
                                                             vmix, 64, 3 * N_NODES, 32, 0);
        k_uin<<<(N_NODES * 160 + 255) / 256, 256, 0, stream>>>(s_cur, vmix, U_in);
        if (p.Wu1) {
            k_gemm<<<dim3(N_NODES / 16, 8), 32, 0, stream>>>(U_in, 160, p.Wu1, 128, p.bu1,
                                                             Hu, 128, N_NODES, 160, 1);
            k_gemm<<<dim3(N_NODES / 16, 10), 32, 0, stream>>>(Hu, 128, p.Wu2, 160, p.bu2,
                                                              Uout, 160, N_NODES, 128, 0);
        } else {
            k_gemm<<<dim3(N_NODES / 16, 10), 32, 0, stream>>>(U_in, 160, p.Wu, 160, p.bu,
                                                              Uout, 160, N_NODES, 160, 0);
        }
        k_final<<<(N_NODES * 224 + 255) / 256, 256, 0, stream>>>(Uout, vmix, s_cur, v_cur);
    }
    (void)in_sizes; (void)n_in; (void)out_size; (void)ws_size;
}